// GCN_90220083019929
// MI455X (gfx1250) — compile-verified
//
#include <hip/hip_runtime.h>
#include <math.h>

typedef __attribute__((ext_vector_type(2))) float v2f;
typedef __attribute__((ext_vector_type(8))) float v8f;

#define IN_FEAT 512
#define HID 16

// ---------------------------------------------------------------------------
// init: zero layer-1 aggregator and d_out (layer-2 aggregator), deg = 1 (self-loop)
// ---------------------------------------------------------------------------
__global__ __launch_bounds__(256) void gcn_init(float* __restrict__ agg,
                                                float* __restrict__ out,
                                                float* __restrict__ deg, int N) {
  size_t gid = (size_t)blockIdx.x * blockDim.x + threadIdx.x;
  size_t n16 = (size_t)N * HID;
  if (gid < n16) { agg[gid] = 0.0f; out[gid] = 0.0f; }
  if (gid < (size_t)N) deg[gid] = 1.0f;  // self-loop contribution
}

// deg[dst] += 1 over real edges
__global__ __launch_bounds__(256) void gcn_degree(const long long* __restrict__ dst,
                                                  float* __restrict__ deg, long long E) {
  long long e = (long long)blockIdx.x * blockDim.x + threadIdx.x;
  if (e < E) atomicAdd(&deg[dst[e]], 1.0f);
}

// deg -> deg^-1/2 in place (deg >= 1 always, matches reference's deg>0 branch)
__global__ __launch_bounds__(256) void gcn_rsqrt(float* __restrict__ deg, int N) {
  int i = blockIdx.x * blockDim.x + threadIdx.x;
  if (i < N) deg[i] = rsqrtf(deg[i]);
}

// ---------------------------------------------------------------------------
// GEMM1: h1[N,16] = x[N,512] @ W1[512,16] using V_WMMA_F32_16X16X4_F32.
// One wave per 16-row tile. W1 staged in LDS (32 KB). Requires N % 16 == 0
// (true: 100000 = 6250*16). Early-out is wave-uniform -> EXEC all ones at WMMA.
//
// K-remap: per outer step of 8, lanes 0-15 supply K = kk..kk+3, lanes 16-31
// supply K = kk+4..kk+7 (one float4 load each); B operand matches that K order.
// ---------------------------------------------------------------------------
__global__ __launch_bounds__(256) void gcn_gemm1(const float* __restrict__ x,
                                                 const float* __restrict__ W1,
                                                 float* __restrict__ h1, int N) {
  __shared__ float lw[IN_FEAT * HID];
  for (int i = threadIdx.x; i < IN_FEAT * HID; i += 256) lw[i] = W1[i];
  __syncthreads();

  const int wave = threadIdx.x >> 5;
  const int lane = threadIdx.x & 31;
  const int half = lane >> 4;       // 0: lanes 0-15, 1: lanes 16-31
  const int m    = lane & 15;
  const int tile = blockIdx.x * 8 + wave;
  if (tile * 16 >= N) return;       // wave-uniform

  const float* xrow = x + (size_t)(tile * 16 + m) * IN_FEAT + 4 * half;
  v8f acc = {};
  for (int kk = 0; kk < IN_FEAT; kk += 8) {
    __builtin_prefetch(xrow + kk + 64, 0, 1);           // global_prefetch_b8
    float4 a4 = *(const float4*)(xrow + kk);            // global_load_b128
    const float* wb = &lw[(kk + 4 * half) * HID + m];   // ds loads
    v2f a0 = {a4.x, a4.y};
    v2f b0 = {wb[0 * HID], wb[1 * HID]};
    acc = __builtin_amdgcn_wmma_f32_16x16x4_f32(false, a0, false, b0,
                                                (short)0, acc, false, false);
    v2f a1 = {a4.z, a4.w};
    v2f b1 = {wb[2 * HID], wb[3 * HID]};
    acc = __builtin_amdgcn_wmma_f32_16x16x4_f32(false, a1, false, b1,
                                                (short)0, acc, false, false);
  }
  // C/D layout: VGPR v holds row M = v + 8*half, col N = m
  float* o = h1 + (size_t)(tile * 16 + 8 * half) * HID + m;
#pragma unroll
  for (int v = 0; v < 8; ++v) o[v * HID] = acc[v];
}

// ---------------------------------------------------------------------------
// Scatter: agg[dst] += (dis[src]*dis[dst]) * h[src], over E real edges plus N
// self-loops (e >= E -> src = dst = e-E). 16 lanes per edge (one per channel):
// coalesced 64B gathers, global_atomic_add_f32 accumulation (L2-resident).
// ---------------------------------------------------------------------------
__global__ __launch_bounds__(256) void gcn_scatter(const float* __restrict__ h,
                                                   const long long* __restrict__ src,
                                                   const long long* __restrict__ dstv,
                                                   const float* __restrict__ dis,
                                                   float* __restrict__ agg,
                                                   long long E, int N) {
  size_t gid = (size_t)blockIdx.x * blockDim.x + threadIdx.x;
  long long e = (long long)(gid >> 4);
  int c = (int)(gid & 15);
  if (e >= E + (long long)N) return;
  long long s, d;
  if (e < E) { s = src[e]; d = dstv[e]; }
  else       { s = d = e - E; }
  float nrm = dis[s] * dis[d];
  atomicAdd(&agg[d * HID + c], nrm * h[s * HID + c]);
}

// agg = relu(agg + b1), in place
__global__ __launch_bounds__(256) void gcn_relu_bias(float* __restrict__ agg,
                                                     const float* __restrict__ b, int N) {
  size_t gid = (size_t)blockIdx.x * blockDim.x + threadIdx.x;
  if (gid < (size_t)N * HID) agg[gid] = fmaxf(agg[gid] + b[gid & 15], 0.0f);
}

// ---------------------------------------------------------------------------
// GEMM2: h2[N,16] = hmid[N,16] @ W2[16,16], K=16 -> 4 WMMA f32 steps.
// ---------------------------------------------------------------------------
__global__ __launch_bounds__(256) void gcn_gemm2(const float* __restrict__ hmid,
                                                 const float* __restrict__ W2,
                                                 float* __restrict__ h2, int N) {
  __shared__ float lw[HID * HID];
  if (threadIdx.x < HID * HID) lw[threadIdx.x] = W2[threadIdx.x];
  __syncthreads();

  const int wave = threadIdx.x >> 5;
  const int lane = threadIdx.x & 31;
  const int half = lane >> 4;
  const int m    = lane & 15;
  const int tile = blockIdx.x * 8 + wave;
  if (tile * 16 >= N) return;       // wave-uniform

  const float* arow = hmid + (size_t)(tile * 16 + m) * HID + 4 * half;
  v8f acc = {};
#pragma unroll
  for (int kk = 0; kk < HID; kk += 8) {
    float4 a4 = *(const float4*)(arow + kk);
    const float* wb = &lw[(kk + 4 * half) * HID + m];
    v2f a0 = {a4.x, a4.y};
    v2f b0 = {wb[0 * HID], wb[1 * HID]};
    acc = __builtin_amdgcn_wmma_f32_16x16x4_f32(false, a0, false, b0,
                                                (short)0, acc, false, false);
    v2f a1 = {a4.z, a4.w};
    v2f b1 = {wb[2 * HID], wb[3 * HID]};
    acc = __builtin_amdgcn_wmma_f32_16x16x4_f32(false, a1, false, b1,
                                                (short)0, acc, false, false);
  }
  float* o = h2 + (size_t)(tile * 16 + 8 * half) * HID + m;
#pragma unroll
  for (int v = 0; v < 8; ++v) o[v * HID] = acc[v];
}

// ---------------------------------------------------------------------------
// out = log_softmax(out + b2) row-wise, in place. One thread per node row.
// ---------------------------------------------------------------------------
__global__ __launch_bounds__(256) void gcn_logsoftmax(float* __restrict__ out,
                                                      const float* __restrict__ b2, int N) {
  int i = blockIdx.x * blockDim.x + threadIdx.x;
  if (i >= N) return;
  float v[HID];
  float4* base = (float4*)(out + (size_t)i * HID);
#pragma unroll
  for (int q = 0; q < 4; ++q) {
    float4 t = base[q];
    v[4 * q + 0] = t.x + b2[4 * q + 0];
    v[4 * q + 1] = t.y + b2[4 * q + 1];
    v[4 * q + 2] = t.z + b2[4 * q + 2];
    v[4 * q + 3] = t.w + b2[4 * q + 3];
  }
  float mx = v[0];
#pragma unroll
  for (int c = 1; c < HID; ++c) mx = fmaxf(mx, v[c]);
  float sum = 0.0f;
#pragma unroll
  for (int c = 0; c < HID; ++c) sum += __expf(v[c] - mx);
  float ls = mx + __logf(sum);
#pragma unroll
  for (int q = 0; q < 4; ++q) {
    float4 t;
    t.x = v[4 * q + 0] - ls;
    t.y = v[4 * q + 1] - ls;
    t.z = v[4 * q + 2] - ls;
    t.w = v[4 * q + 3] - ls;
    base[q] = t;
  }
}

// ---------------------------------------------------------------------------
// Launch. Inputs: x[N,512] f32, edge_index[2,E] i64, W1[512,16], b1[16],
// W2[16,16], b2[16]. Output: [N,16] f32 log-probs.
// Workspace (floats): dis[N] | h1/h2[16N] | agg1[16N]  (~13.2 MB)
// ---------------------------------------------------------------------------
extern "C" void kernel_launch(void* const* d_in, const int* in_sizes, int n_in,
                              void* d_out, int out_size, void* d_ws, size_t ws_size,
                              hipStream_t stream) {
  const float*     x  = (const float*)d_in[0];
  const long long* ei = (const long long*)d_in[1];   // int64 edge_index
  const float*     W1 = (const float*)d_in[2];
  const float*     b1 = (const float*)d_in[3];
  const float*     W2 = (const float*)d_in[4];
  const float*     b2 = (const float*)d_in[5];
  float* out = (float*)d_out;

  const int       N = in_sizes[0] / IN_FEAT;   // 100000
  const long long E = in_sizes[1] / 2;         // 3200000
  const long long* src = ei;
  const long long* dst = ei + E;

  float* ws   = (float*)d_ws;
  float* dis  = ws;                            // N floats (deg, then deg^-1/2)
  float* h1   = ws + (size_t)N;                // 16N floats (layer-1 h, reused for h2)
  float* agg1 = ws + (size_t)17 * N;           // 16N floats (layer-1 aggregate)

  const size_t n16   = (size_t)N * HID;
  const unsigned bN16 = (unsigned)((n16 + 255) / 256);
  const unsigned bN   = (unsigned)((N + 255) / 256);
  const unsigned bE   = (unsigned)((E + 255) / 256);
  const unsigned nTiles = (unsigned)((N + 15) / 16);
  const unsigned bGemm  = (nTiles + 7) / 8;    // 8 waves (tiles) per 256-thread block
  const size_t scTot    = (size_t)(E + N) * HID;
  const unsigned bScat  = (unsigned)((scTot + 255) / 256);

  gcn_init<<<bN16, 256, 0, stream>>>(agg1, out, dis, N);
  gcn_degree<<<bE, 256, 0, stream>>>(dst, dis, E);
  gcn_rsqrt<<<bN, 256, 0, stream>>>(dis, N);

  gcn_gemm1<<<bGemm, 256, 0, stream>>>(x, W1, h1, N);
  gcn_scatter<<<bScat, 256, 0, stream>>>(h1, src, dst, dis, agg1, E, N);
  gcn_relu_bias<<<bN16, 256, 0, stream>>>(agg1, b1, N);

  gcn_gemm2<<<bGemm, 256, 0, stream>>>(agg1, W2, h1, N);   // h1 reused as h2
  gcn_scatter<<<bScat, 256, 0, stream>>>(h1, src, dst, dis, out, E, N);
  gcn_logsoftmax<<<bN, 256, 0, stream>>>(out, b2, N);
}